// AttentionLayer_30021821399887
// MI455X (gfx1250) — compile-verified
//
#include <hip/hip_runtime.h>

// ---------------------------------------------------------------------------
// AttentionLayer for MI455X (gfx1250): f32 WMMA 16x16x4 everywhere.
//   q/k/v = X @ W^T + b        (gemm64x64: async-to-LDS staged W tile)
//   flash-style masked softmax attention per (b, h)   (flash_attn)
//   out   = attn @ Wo^T + bo   (gemm64x64)
// ---------------------------------------------------------------------------

typedef float v2f __attribute__((ext_vector_type(2)));
typedef float v8f __attribute__((ext_vector_type(8)));

#define D_MODEL 1024
#define N_HEADS 16
#define HEAD_E  64
#define SEQ_L   2048
#define BATCH   2
#define M_TOTAL (BATCH * SEQ_L)

__device__ __forceinline__ v8f wmma_f32(v2f a, v2f b, v8f c) {
  // D = A(16x4, f32) x B(4x16, f32) + C(16x16, f32)
  return __builtin_amdgcn_wmma_f32_16x16x4_f32(
      /*neg_a=*/false, a, /*neg_b=*/false, b,
      /*c_mod=*/(short)0, c, /*reuse_a=*/false, /*reuse_b=*/false);
}

// CDNA5 async copy: global memory -> LDS, 16B per lane, tracked by ASYNCcnt.
// lds_off must be the real LDS byte offset (low 32 bits of the generic
// pointer) so the shared array is captured and alias analysis keeps the
// subsequent ds_load reads live.
__device__ __forceinline__ void async_load_b128(unsigned lds_off, const float* gptr) {
  asm volatile("global_load_async_to_lds_b128 %0, %1, off"
               :: "v"(lds_off), "v"(gptr) : "memory");
}
__device__ __forceinline__ void wait_asynccnt0() {
  asm volatile("s_wait_asynccnt 0x0" ::: "memory");
}

__device__ __forceinline__ float rmax16(float x) {
  // reduce over the 16 lanes of each wave half (xor masks < 16 stay in-half)
  x = fmaxf(x, __shfl_xor(x, 1));
  x = fmaxf(x, __shfl_xor(x, 2));
  x = fmaxf(x, __shfl_xor(x, 4));
  x = fmaxf(x, __shfl_xor(x, 8));
  return x;
}
__device__ __forceinline__ float rsum16(float x) {
  x += __shfl_xor(x, 1);
  x += __shfl_xor(x, 2);
  x += __shfl_xor(x, 4);
  x += __shfl_xor(x, 8);
  return x;
}

// ---------------------------------------------------------------------------
// C[M,N] = X[M,K] @ W[N,K]^T + bias[N]
// Workgroup = 4 waves -> 64x64 C tile (each wave: 16 rows of M, all 64 N).
// Per 32-wide k-slice: the 64x32 W sub-tile (8 KB) is staged into LDS with
// GLOBAL_LOAD_ASYNC_TO_LDS_B128 (each wave copies 1/4), fenced with
// s_wait_asynccnt + barrier, then consumed as ds_load b64 B-fragments.
// A layout: lane m = l&15, vgpr v -> k = 2*(l>>4)+v   (float2 along K)
// D layout: lane n = l&15, vgpr v -> m = v + 8*(l>>4)
// ---------------------------------------------------------------------------
#define KSLICE 32

__global__ __launch_bounds__(128)
void gemm64x64(const float* __restrict__ X, const float* __restrict__ W,
               const float* __restrict__ bias, float* __restrict__ C) {
  __shared__ float lds_w[64 * KSLICE];   // [n-in-tile][k-in-slice], 8 KB

  const int K = D_MODEL, N = D_MODEL;
  const int lane = threadIdx.x & 31;
  const int wave = threadIdx.x >> 5;     // 0..3
  const int nn = lane & 15, half = lane >> 4;
  const int m0 = blockIdx.x * 64 + wave * 16;
  const int n0 = blockIdx.y * 64;

  // staging assignment: 512 16B-chunks per slice; this lane copies 4
  // consecutive chunks = 16 floats of one W row.
  const int g0   = wave * 128 + lane * 4;        // first chunk id
  const int srow = g0 >> 3;                      // 0..63 (n within tile)
  const int scol = (g0 & 7) * 4;                 // k offset within slice
  const float* wsrc_base = W + (size_t)(n0 + srow) * K + scol;
  // Low 32 bits of the generic pointer == LDS byte offset (aperture mapping);
  // using the real address also captures lds_w for the asm's memory clobber.
  const unsigned ldst_base =
      (unsigned)(size_t)(&lds_w[0]) + (unsigned)(srow * KSLICE + scol) * 4u;

  const float* xr = X + (size_t)(m0 + nn) * K + 2 * half;

  v8f acc0 = {}, acc1 = {}, acc2 = {}, acc3 = {};

  for (int ks = 0; ks < K; ks += KSLICE) {
    __syncthreads();                     // previous slice fully consumed
    // ---- async stage W[n0..n0+63][ks..ks+31] -> LDS ----
    const float* wsrc = wsrc_base + ks;
    #pragma unroll
    for (int t = 0; t < 4; ++t)
      async_load_b128(ldst_base + 16u * t, wsrc + 4 * t);
    // prefetch next A-slice while the async copy is in flight
    if (ks + KSLICE < K) __builtin_prefetch(xr + ks + KSLICE, 0, 3);
    wait_asynccnt0();
    __syncthreads();                     // all waves' chunks visible

    // ---- consume: 8 wmma k-steps of 4 ----
    #pragma unroll
    for (int k = 0; k < KSLICE; k += 4) {
      v2f a = *(const v2f*)(xr + ks + k);
      v2f b0 = *(const v2f*)&lds_w[( 0 + nn) * KSLICE + k + 2 * half];
      v2f b1 = *(const v2f*)&lds_w[(16 + nn) * KSLICE + k + 2 * half];
      v2f b2 = *(const v2f*)&lds_w[(32 + nn) * KSLICE + k + 2 * half];
      v2f b3 = *(const v2f*)&lds_w[(48 + nn) * KSLICE + k + 2 * half];
      acc0 = wmma_f32(a, b0, acc0);
      acc1 = wmma_f32(a, b1, acc1);
      acc2 = wmma_f32(a, b2, acc2);
      acc3 = wmma_f32(a, b3, acc3);
    }
  }

  const float b0 = bias[n0 +  0 + nn];
  const float b1 = bias[n0 + 16 + nn];
  const float b2 = bias[n0 + 32 + nn];
  const float b3 = bias[n0 + 48 + nn];

  #pragma unroll
  for (int v = 0; v < 8; ++v) {
    const int m = m0 + v + 8 * half;
    float* crow = C + (size_t)m * N + n0 + nn;
    crow[0]  = acc0[v] + b0;
    crow[16] = acc1[v] + b1;
    crow[32] = acc2[v] + b2;
    crow[48] = acc3[v] + b3;
  }
}

// ---------------------------------------------------------------------------
// Flash-style attention for one (b, h, 16-query block) per wave.
// Mask (anti-causal): key j visible to query i iff (j >= i + d) || (j == L-1),
// d = 2^(h / (H/DIAG)) = 2^(h>>2).  Blocks kb < qb are fully masked -> skipped.
// ---------------------------------------------------------------------------
__global__ __launch_bounds__(32)
void flash_attn(const float* __restrict__ Q, const float* __restrict__ Kmat,
                const float* __restrict__ V, float* __restrict__ O) {
  __shared__ float pbuf[16 * 18];  // P tile staging, padded rows (8B-aligned pairs)

  const int lane = threadIdx.x;
  const int nn = lane & 15, half = lane >> 4;
  const int qb = blockIdx.x;           // query block 0..127
  const int h  = blockIdx.y;           // head 0..15
  const int b  = blockIdx.z;           // batch 0..1
  const int q0 = qb * 16;
  const int d  = 1 << (h >> 2);        // diag offset for this head's group
  const size_t base = (size_t)b * SEQ_L * D_MODEL + (size_t)h * HEAD_E;
  const float scale = 0.125f;          // 1/sqrt(64)

  // Q tile in A-layout register chunks: qa[ec][v] = Q[q0 + (l&15)][4*ec + 2*half + v]
  v2f qa[16];
  {
    const float* qr = Q + base + (size_t)(q0 + nn) * D_MODEL + 2 * half;
    #pragma unroll
    for (int ec = 0; ec < 16; ++ec) qa[ec] = *(const v2f*)(qr + ec * 4);
  }

  v8f o0 = {}, o1 = {}, o2 = {}, o3 = {};
  float rm[8], rs[8];
  #pragma unroll
  for (int v = 0; v < 8; ++v) { rm[v] = -3.0e38f; rs[v] = 0.0f; }

  for (int kb = qb; kb < SEQ_L / 16; ++kb) {
    const int kbase = kb * 16;

    // ---- S = Q @ K^T over E=64 (16 wmma steps of k=4) ----
    v8f s = {};
    const float* kr = Kmat + base + (size_t)(kbase + nn) * D_MODEL + 2 * half;
    #pragma unroll
    for (int ec = 0; ec < 16; ++ec)
      s = wmma_f32(qa[ec], *(const v2f*)(kr + ec * 4), s);

    // ---- mask + scale + online softmax (row stats via in-half shuffles) ----
    float p[8], alpha[8];
    #pragma unroll
    for (int v = 0; v < 8; ++v) {
      const int i  = q0 + v + 8 * half;   // query row of this D slot
      const int jk = kbase + nn;          // key column of this lane
      const bool ok = (jk >= i + d) || (jk == SEQ_L - 1);
      const float sm = ok ? s[v] * scale : -3.0e38f;
      const float nm = fmaxf(rm[v], rmax16(sm));
      alpha[v] = __expf(rm[v] - nm);
      rm[v] = nm;
      const float pv = ok ? __expf(sm - nm) : 0.0f;  // explicit 0 for masked lanes
      p[v] = pv;
      rs[v] = rs[v] * alpha[v] + rsum16(pv);
    }
    #pragma unroll
    for (int v = 0; v < 8; ++v) {
      o0[v] *= alpha[v]; o1[v] *= alpha[v]; o2[v] *= alpha[v]; o3[v] *= alpha[v];
    }

    // ---- stage P (D-layout) -> LDS -> A-layout for the PV wmmas ----
    __syncthreads();
    #pragma unroll
    for (int v = 0; v < 8; ++v) pbuf[(v + 8 * half) * 18 + nn] = p[v];
    __syncthreads();

    const float* vr = V + base + (size_t)kbase * D_MODEL + nn;
    #pragma unroll
    for (int k0 = 0; k0 < 16; k0 += 4) {
      v2f ap = *(const v2f*)&pbuf[nn * 18 + k0 + 2 * half];
      const int row = k0 + 2 * half;
      v2f b0v = { vr[(size_t)row * D_MODEL +  0], vr[(size_t)(row + 1) * D_MODEL +  0] };
      v2f b1v = { vr[(size_t)row * D_MODEL + 16], vr[(size_t)(row + 1) * D_MODEL + 16] };
      v2f b2v = { vr[(size_t)row * D_MODEL + 32], vr[(size_t)(row + 1) * D_MODEL + 32] };
      v2f b3v = { vr[(size_t)row * D_MODEL + 48], vr[(size_t)(row + 1) * D_MODEL + 48] };
      o0 = wmma_f32(ap, b0v, o0);
      o1 = wmma_f32(ap, b1v, o1);
      o2 = wmma_f32(ap, b2v, o2);
      o3 = wmma_f32(ap, b3v, o3);
    }
  }

  // ---- finalize: O /= rowsum, write [b, q, h*64 + e] ----
  #pragma unroll
  for (int v = 0; v < 8; ++v) {
    const float inv = 1.0f / rs[v];
    float* orow = O + base + (size_t)(q0 + v + 8 * half) * D_MODEL + nn;
    orow[0]  = o0[v] * inv;
    orow[16] = o1[v] * inv;
    orow[32] = o2[v] * inv;
    orow[48] = o3[v] * inv;
  }
}

// ---------------------------------------------------------------------------
extern "C" void kernel_launch(void* const* d_in, const int* in_sizes, int n_in,
                              void* d_out, int out_size, void* d_ws, size_t ws_size,
                              hipStream_t stream) {
  const float* queries = (const float*)d_in[0];
  const float* keys    = (const float*)d_in[1];
  const float* values  = (const float*)d_in[2];
  const float* Wq = (const float*)d_in[3];
  const float* bq = (const float*)d_in[4];
  const float* Wk = (const float*)d_in[5];
  const float* bk = (const float*)d_in[6];
  const float* Wv = (const float*)d_in[7];
  const float* bv = (const float*)d_in[8];
  const float* Wo = (const float*)d_in[9];
  const float* bo = (const float*)d_in[10];

  const size_t plane = (size_t)M_TOTAL * D_MODEL;  // 4096*1024 floats = 16 MB
  float* qws = (float*)d_ws;
  float* kws = qws + plane;
  float* vws = kws + plane;
  float* aws = vws + plane;

  dim3 gg(M_TOTAL / 64, D_MODEL / 64, 1);   // 64 x 16 workgroups, 4 waves each
  gemm64x64<<<gg, 128, 0, stream>>>(queries, Wq, bq, qws);
  gemm64x64<<<gg, 128, 0, stream>>>(keys,    Wk, bk, kws);
  gemm64x64<<<gg, 128, 0, stream>>>(values,  Wv, bv, vws);

  dim3 ga(SEQ_L / 16, N_HEADS, BATCH);      // 128 x 16 x 2 waves
  flash_attn<<<ga, 32, 0, stream>>>(qws, kws, vws, aws);

  gemm64x64<<<gg, 128, 0, stream>>>(aws, Wo, bo, (float*)d_out);
}